// MoreauTropicalKernel_45947560133335
// MI455X (gfx1250) — compile-verified
//
#include <hip/hip_runtime.h>
#include <hip/hip_bf16.h>
#include <stdint.h>

// Moreau-tropical forward on gfx1250.
// One wave32 per (b, n) output. z = x[b,:] + W[n,:] (D=1024) lives in
// registers: 8 x float4 per lane. 32 bisection steps use the identity
//   sum_d relu(z_d - mid) = sum_d max(z_d, mid) - D*mid
// => 2 VALU ops per element in the hot loop. Cross-lane reduction via
// shfl_xor butterfly (wave32). x row is staged into LDS once per block
// with the CDNA5 async global->LDS path (ASYNCcnt).

#define D_FIXED 1024
#define WPB 8                  // waves per block (block shares one b)
#define TPB (WPB * 32)

#if defined(__HIP_DEVICE_COMPILE__)
#if __has_builtin(__builtin_amdgcn_global_load_async_to_lds_b128)
#define USE_ASYNC_LDS 1
#endif
#endif

// Pointee type demanded by the builtin per hipcc diagnostic:
// 'int __attribute__((__vector_size__(4 * sizeof(int))))'
typedef int v4i __attribute__((vector_size(4 * sizeof(int))));
typedef __attribute__((address_space(1))) v4i* gv4i_ptr;
typedef __attribute__((address_space(3))) v4i* lv4i_ptr;

__device__ __forceinline__ float wave_sum(float v) {
#pragma unroll
  for (int m = 16; m >= 1; m >>= 1) v += __shfl_xor(v, m, 32);
  return v;
}
__device__ __forceinline__ float wave_min(float v) {
#pragma unroll
  for (int m = 16; m >= 1; m >>= 1) v = fminf(v, __shfl_xor(v, m, 32));
  return v;
}
__device__ __forceinline__ float wave_max(float v) {
#pragma unroll
  for (int m = 16; m >= 1; m >>= 1) v = fmaxf(v, __shfl_xor(v, m, 32));
  return v;
}

__global__ __launch_bounds__(TPB)
void moreau_tropical_fwd(const float* __restrict__ x,
                         const float* __restrict__ W,
                         const float* __restrict__ lam_p,
                         float* __restrict__ out,
                         int B, int N) {
  __shared__ float xs[D_FIXED];

  const int lane = threadIdx.x & 31;
  const int wv   = threadIdx.x >> 5;
  const int b    = blockIdx.y;
  const int n    = blockIdx.x * WPB + wv;
  const float lam = lam_p[0];

  // ---- Stage x[b, :] into LDS: 256 threads x 16B = 4 KB (async path) ----
  {
    const float* gsrc = x + (size_t)b * D_FIXED + (size_t)threadIdx.x * 4;
    float* ldst = &xs[threadIdx.x * 4];
#if defined(USE_ASYNC_LDS)
    __builtin_amdgcn_global_load_async_to_lds_b128(
        (gv4i_ptr)gsrc, (lv4i_ptr)ldst, /*offset=*/0, /*cpol=*/0);
#if __has_builtin(__builtin_amdgcn_s_wait_asynccnt)
    __builtin_amdgcn_s_wait_asynccnt(0);
#else
    asm volatile("s_wait_asynccnt 0" ::: "memory");
#endif
#else
    *(float4*)ldst = *(const float4*)gsrc;
#endif
  }
  __syncthreads();

  if (n >= N) return;

  // ---- Build z = W[n,:] + x[b,:]; lane holds d = j*128 + lane*4 + k ----
  const float4* __restrict__ Wr = (const float4*)(W + (size_t)n * D_FIXED);
  const float4* Xr = (const float4*)xs;
  float4 z[8];
#pragma unroll
  for (int j = 0; j < 8; ++j) {
    float4 wvv = Wr[j * 32 + lane];
    float4 xvv = Xr[j * 32 + lane];
    z[j].x = wvv.x + xvv.x;
    z[j].y = wvv.y + xvv.y;
    z[j].z = wvv.z + xvv.z;
    z[j].w = wvv.w + xvv.w;
  }

  // ---- tau bounds: lo = min(z) - lam, hi = max(z) ----
  float mn = z[0].x, mx = z[0].x;
#pragma unroll
  for (int j = 0; j < 8; ++j) {
    mn = fminf(mn, fminf(fminf(z[j].x, z[j].y), fminf(z[j].z, z[j].w)));
    mx = fmaxf(mx, fmaxf(fmaxf(z[j].x, z[j].y), fmaxf(z[j].z, z[j].w)));
  }
  mn = wave_min(mn);
  mx = wave_max(mx);
  float lo = mn - lam;
  float hi = mx;

  // ---- 32 bisection steps ----
#pragma unroll 1
  for (int it = 0; it < 32; ++it) {
    const float mid = 0.5f * (lo + hi);
    // sum_d max(z_d, mid); 4 independent accumulators for dual-issue
    float a0 = 0.f, a1 = 0.f, a2 = 0.f, a3 = 0.f;
#pragma unroll
    for (int j = 0; j < 8; ++j) {
      a0 += fmaxf(z[j].x, mid);
      a1 += fmaxf(z[j].y, mid);
      a2 += fmaxf(z[j].z, mid);
      a3 += fmaxf(z[j].w, mid);
    }
    float s = wave_sum((a0 + a1) + (a2 + a3));               // lane-uniform
    const float f = __builtin_fmaf(-(float)D_FIXED, mid, s); // f(mid)
    const bool too_low = f > lam;
    lo = too_low ? mid : lo;
    hi = too_low ? hi : mid;
  }
  const float tau = 0.5f * (lo + hi);

  // ---- delta = sum relu(z - tau)^2 / (2*lam); y = tau + delta ----
  float d0 = 0.f, d1 = 0.f, d2 = 0.f, d3 = 0.f;
#pragma unroll
  for (int j = 0; j < 8; ++j) {
    float t0 = fmaxf(z[j].x - tau, 0.f); d0 = __builtin_fmaf(t0, t0, d0);
    float t1 = fmaxf(z[j].y - tau, 0.f); d1 = __builtin_fmaf(t1, t1, d1);
    float t2 = fmaxf(z[j].z - tau, 0.f); d2 = __builtin_fmaf(t2, t2, d2);
    float t3 = fmaxf(z[j].w - tau, 0.f); d3 = __builtin_fmaf(t3, t3, d3);
  }
  const float sq = wave_sum((d0 + d1) + (d2 + d3));
  const float y = tau + sq / (2.0f * lam);

  if (lane == 0) out[(size_t)b * N + n] = y;
}

extern "C" void kernel_launch(void* const* d_in, const int* in_sizes, int n_in,
                              void* d_out, int out_size, void* d_ws, size_t ws_size,
                              hipStream_t stream) {
  const float* x   = (const float*)d_in[0];   // [B, D] fp32
  const float* W   = (const float*)d_in[1];   // [N, D] fp32
  const float* lam = (const float*)d_in[2];   // scalar fp32
  float* out = (float*)d_out;                 // [B, N] fp32

  const int B = in_sizes[0] / D_FIXED;
  const int N = in_sizes[1] / D_FIXED;

  dim3 grid((N + WPB - 1) / WPB, B);
  dim3 block(TPB);
  hipLaunchKernelGGL(moreau_tropical_fwd, grid, block, 0, stream,
                     x, W, lam, out, B, N);
}